// T2FN_33363305955371
// MI455X (gfx1250) — compile-verified
//
#include <hip/hip_runtime.h>
#include <hip/hip_bf16.h>

typedef __attribute__((ext_vector_type(2))) float v2f;
typedef __attribute__((ext_vector_type(8))) float v8f;

// explicit global-memory pointers so selects don't degrade to FLAT loads
typedef const float __attribute__((address_space(1)))* gcf;
typedef const v2f   __attribute__((address_space(1)))* gv2f;

#define B_    128
#define S_    128
#define AH_   64
#define VH_   64
#define TH_   128
#define AD    65            // audio dim + ones
#define VD    65
#define WD    129           // text dim + ones
#define AV_   (AD * VD)     // 4225
#define FUSED 545025        // 65*65*129
#define PF_   128
#define NAVT  265           // ceil(4225/16)
#define CH_STRIDE 518       // 516 f-values + pad; even -> b64-aligned LDS pairs,
                            // stride%64 = 6 -> 16 lanes hit 16 distinct banks

// ---------------------------------------------------------------------------
// ws layout (floats):
//   [0 .. 16384)      y1partial [B][PF]  (atomically accumulated, zeroed first)
//   [16384 .. 16512)  per-batch norm  sqrt(||fusion_b||^2)
//   [16512]           1.0f   (broadcast source for "ones" virtual column)
//   [16513]           0.0f   (broadcast source for padded lanes)
// ---------------------------------------------------------------------------

__global__ void init_ws(float* ws) {
    int i = blockIdx.x * 256 + threadIdx.x;
    if (i < B_ * PF_) ws[i] = 0.0f;
    if (i == 0) { ws[16512] = 1.0f; ws[16513] = 0.0f; }
}

// ---------------------------------------------------------------------------
// Fused kernel: builds fusion tiles (stage 1) and contracts with W1 (stage 2)
// without ever materializing fusion in global memory.
//   grid = 265 av-tiles (16 rows of (a,v) each), block = 256 threads = 8 waves
//   stage 1: per wave, 2 batches:  C1[16av,16w] += P[16av,4s] x T[4s,16w]
//   stage 2: per wave, 1 p-tile:   C2[16p,16b]  += W1[16p,4f] x chunk^T[4f,16b]
// All inner-loop loads are unconditional: lanes that need the constant "1"
// column (or zero padding) read through stride-0 pointers into ws constants.
// ---------------------------------------------------------------------------
__global__ __launch_bounds__(256)
void fused_fusion_w1(const float* __restrict__ audio,
                     const float* __restrict__ vision,
                     const float* __restrict__ text,
                     const float* __restrict__ W1,
                     float* __restrict__ y1p,
                     const float* __restrict__ ones,
                     const float* __restrict__ zeros)
{
    __shared__ float chunk[16 * CH_STRIDE];   // [b_local][f_sub 0..515]

    const int tid  = threadIdx.x;
    const int wave = tid >> 5;
    const int lane = tid & 31;
    const int l16  = lane & 15;
    const bool hi  = lane >= 16;
    const int offs = hi ? 2 : 0;            // first K element of this lane half

    const int avtile = blockIdx.x;          // 0..264
    const int avbase = avtile * 16;
    const bool tail  = (avtile == NAVT - 1);

    // per-lane (a,v) decomposition for the phase-1 A operand (M = av row)
    const int  av    = avbase + l16;
    const bool avok  = av < AV_;
    const int  a_i   = avok ? (av / VD) : 0;
    const int  v_i   = avok ? (av % VD) : 0;

    // ---- per-lane factor descriptors (branchless inner loop) --------------
    const bool  dirA = avok && (a_i > 0);
    const bool  dirV = avok && (v_i > 0);
    const gcf specAV = (gcf)(avok ? ones : zeros); // a_i==0/v_i==0 -> 1.0 ; pad -> 0
    const int strA = dirA ? AH_ : 0;
    const int strV = dirV ? VH_ : 0;
    const int aoff0 = offs * strA, aoff1 = aoff0 + strA, astep = 4 * strA;
    const int voff0 = offs * strV, voff1 = voff0 + strV, vstep = 4 * strV;

    bool dirT[9]; gcf specT[9]; int strT[9], toff0[9], toff1[9], tstep[9];
    #pragma unroll
    for (int wt = 0; wt < 9; ++wt) {
        const int w0 = wt * 16 + l16;
        dirT[wt]  = (w0 >= 1) && (w0 <= 128);
        specT[wt] = (gcf)((w0 == 0) ? ones : zeros);
        strT[wt]  = dirT[wt] ? TH_ : 0;
        toff0[wt] = offs * strT[wt];
        toff1[wt] = toff0[wt] + strT[wt];
        tstep[wt] = 4 * strT[wt];
    }

    const size_t fbase = (size_t)avbase * WD;   // first flat f of this tile
    const int    prow  = wave * 16 + l16;       // phase-2 W1 row for this lane
    const gcf    w1row = (gcf)(W1 + (size_t)prow * (size_t)FUSED);

    for (int bt = 0; bt < 8; ++bt) {
        // =================== phase 1: fusion tiles (held in VGPRs) ==========
        v8f c1[2][9];
        #pragma unroll
        for (int i = 0; i < 2; ++i)
            #pragma unroll
            for (int wt = 0; wt < 9; ++wt)
                c1[i][wt] = (v8f){0,0,0,0,0,0,0,0};

        #pragma unroll
        for (int i = 0; i < 2; ++i) {
            const int b = bt * 16 + wave * 2 + i;
            const float* ab = audio  + (size_t)b * S_ * AH_;
            const float* vb = vision + (size_t)b * S_ * VH_;
            const float* tb = text   + (size_t)b * S_ * TH_;

            gcf pA = dirA ? (gcf)(ab + (a_i - 1)) : specAV;
            gcf pV = dirV ? (gcf)(vb + (v_i - 1)) : specAV;
            gcf pT[9];
            #pragma unroll
            for (int wt = 0; wt < 9; ++wt) {
                const int w0 = wt * 16 + l16;
                pT[wt] = dirT[wt] ? (gcf)(tb + (w0 - 1)) : specT[wt];
            }

            for (int kk = 0; kk < 32; ++kk) {
                const float A0 = pA[aoff0], A1 = pA[aoff1];
                const float V0 = pV[voff0], V1 = pV[voff1];
                v2f aop; aop.x = A0 * V0; aop.y = A1 * V1;
                #pragma unroll
                for (int wt = 0; wt < 9; ++wt) {
                    v2f bop; bop.x = pT[wt][toff0[wt]]; bop.y = pT[wt][toff1[wt]];
                    c1[i][wt] = __builtin_amdgcn_wmma_f32_16x16x4_f32(
                        false, aop, false, bop, (short)0, c1[i][wt], false, false);
                    pT[wt] += tstep[wt];
                }
                pA += astep; pV += vstep;
            }
        }

        // ============ sub-slabs: spill 4 av-rows to LDS, contract with W1 ===
        v8f c2 = (v8f){0,0,0,0,0,0,0,0};
        #pragma unroll
        for (int sub = 0; sub < 4; ++sub) {
            __syncthreads();   // previous phase-2 pass done reading chunk
            // store rows r = 4*sub .. 4*sub+3 of this wave's C1 tiles
            #pragma unroll
            for (int i = 0; i < 2; ++i) {
                const int bloc = wave * 2 + i;
                #pragma unroll
                for (int rr = 0; rr < 4; ++rr) {
                    const int r     = sub * 4 + rr;
                    const bool mine = (r < 8) ? (!hi) : hi;   // half holding row r
                    const int  j    = (r < 8) ? r : (r - 8);  // C VGPR index
                    if (mine) {
                        #pragma unroll
                        for (int wt = 0; wt < 9; ++wt) {
                            const int w = wt * 16 + l16;
                            if (w < WD)
                                chunk[bloc * CH_STRIDE + rr * WD + w] = c1[i][wt][j];
                        }
                    }
                }
            }
            __syncthreads();

            // phase 2: K = 516 f-values of this slab, 129 k-steps of 4
            const size_t f0 = fbase + (size_t)sub * 516;
            if (!tail) {
                // prefetch the next slab of this lane's W1 row
                __builtin_prefetch((const float*)0 + (size_t)(w1row + f0 + 516 -
                                   (gcf)0), 0, 1);
                for (int kk = 0; kk < 129; ++kk) {
                    const int fl = kk * 4 + offs;
                    v2f aop = *(gv2f)(w1row + f0 + fl);          // b64 W1 stream
                    v2f bop = *(const v2f*)&chunk[l16 * CH_STRIDE + fl]; // ds b64
                    c2 = __builtin_amdgcn_wmma_f32_16x16x4_f32(
                        false, aop, false, bop, (short)0, c2, false, false);
                }
            } else {
                for (int kk = 0; kk < 129; ++kk) {
                    const int fl = kk * 4 + offs;
                    const size_t fg = f0 + fl;
                    const size_t i0 = (fg     < FUSED) ? fg       : 0;
                    const size_t i1 = (fg + 1 < FUSED) ? (fg + 1) : 0;
                    v2f aop;
                    aop.x = (fg     < FUSED) ? w1row[i0] : 0.f;
                    aop.y = (fg + 1 < FUSED) ? w1row[i1] : 0.f;
                    v2f bop = *(const v2f*)&chunk[l16 * CH_STRIDE + fl];
                    c2 = __builtin_amdgcn_wmma_f32_16x16x4_f32(
                        false, aop, false, bop, (short)0, c2, false, false);
                }
            }
        }

        // scatter-add C2[16p,16b] into y1partial
        {
            const int pbase = wave * 16 + (hi ? 8 : 0);
            const int bcol  = bt * 16 + l16;
            #pragma unroll
            for (int j = 0; j < 8; ++j)
                atomicAdd(&y1p[bcol * PF_ + pbase + j], c2[j]);
        }
    }
}

// ---------------------------------------------------------------------------
// reg_loss via Gram factorization:
//   ||fusion_b||^2 = sum_{s,s'} (1+<a_s,a_s'>)(1+<v_s,v_s'>)(1+<t_s,t_s'>)
// one block per batch; thread owns a fixed s-row (reused 64x from cache)
// ---------------------------------------------------------------------------
__global__ __launch_bounds__(256)
void norm_kernel(const float* __restrict__ audio,
                 const float* __restrict__ vision,
                 const float* __restrict__ text,
                 float* __restrict__ normbuf)
{
    __shared__ float red[256];
    const int b   = blockIdx.x;
    const int tid = threadIdx.x;
    const float* ab = audio  + (size_t)b * S_ * AH_;
    const float* vb = vision + (size_t)b * S_ * VH_;
    const float* tb = text   + (size_t)b * S_ * TH_;

    float acc = 0.f;
    for (int j = 0; j < 64; ++j) {
        const int pi = tid * 64 + j;      // s constant per thread -> L0 reuse
        const int s  = pi >> 7;
        const int s2 = pi & 127;
        float ga = 1.f, gv = 1.f, gt = 1.f;
        for (int k = 0; k < AH_; ++k) ga += ab[s * AH_ + k] * ab[s2 * AH_ + k];
        for (int k = 0; k < VH_; ++k) gv += vb[s * VH_ + k] * vb[s2 * VH_ + k];
        for (int k = 0; k < TH_; ++k) gt += tb[s * TH_ + k] * tb[s2 * TH_ + k];
        acc += ga * gv * gt;
    }
    red[tid] = acc;
    __syncthreads();
    for (int o = 128; o > 0; o >>= 1) {
        if (tid < o) red[tid] += red[tid + o];
        __syncthreads();
    }
    if (tid == 0) normbuf[b] = sqrtf(red[0]);
}

// ---------------------------------------------------------------------------
// MLP tail: y1=relu(y1p+b1); y2=relu(y1 W2^T + b2); out=sigmoid(y2 W3^T+b3)*6-3
// ---------------------------------------------------------------------------
__global__ __launch_bounds__(128)
void mlp_tail(const float* __restrict__ y1p, const float* __restrict__ b1,
              const float* __restrict__ W2,  const float* __restrict__ b2,
              const float* __restrict__ W3,  const float* __restrict__ b3,
              float* __restrict__ out)
{
    __shared__ float y1s[128];
    __shared__ float y2s[128];
    const int b = blockIdx.x, p = threadIdx.x;
    y1s[p] = fmaxf(y1p[b * PF_ + p] + b1[p], 0.f);
    __syncthreads();
    float acc = b2[p];
    for (int q = 0; q < PF_; ++q) acc += y1s[q] * W2[p * PF_ + q];
    y2s[p] = fmaxf(acc, 0.f) * W3[p];
    __syncthreads();
    for (int o = 64; o > 0; o >>= 1) {
        if (p < o) y2s[p] += y2s[p + o];
        __syncthreads();
    }
    if (p == 0) {
        const float z = y2s[0] + b3[0];
        out[b] = 6.f / (1.f + expf(-z)) - 3.f;
    }
}

// fixed-order reduction so reg_loss is bit-deterministic
__global__ void reg_reduce(const float* __restrict__ normbuf, float* __restrict__ out)
{
    if (threadIdx.x == 0 && blockIdx.x == 0) {
        float s = 0.f;
        for (int i = 0; i < B_; ++i) s += normbuf[i];
        // tmp = sqrt(64*64*128/128) = 64 ; mean over B=128 -> 0.5 * sum
        out[B_] = 0.5f * s;
    }
}

extern "C" void kernel_launch(void* const* d_in, const int* in_sizes, int n_in,
                              void* d_out, int out_size, void* d_ws, size_t ws_size,
                              hipStream_t stream)
{
    (void)in_sizes; (void)n_in; (void)out_size; (void)ws_size;
    const float* audio  = (const float*)d_in[0];
    const float* vision = (const float*)d_in[1];
    const float* text   = (const float*)d_in[2];
    const float* W1     = (const float*)d_in[3];
    const float* b1     = (const float*)d_in[4];
    const float* W2     = (const float*)d_in[5];
    const float* b2     = (const float*)d_in[6];
    const float* W3     = (const float*)d_in[7];
    const float* b3     = (const float*)d_in[8];

    float* out     = (float*)d_out;            // [0..127]=out, [128]=reg_loss
    float* ws      = (float*)d_ws;
    float* y1p     = ws;                       // 16384 floats
    float* normbuf = ws + B_ * PF_;            // 128 floats
    const float* ones  = ws + 16512;
    const float* zeros = ws + 16513;

    init_ws        <<<64,   256, 0, stream>>>(ws);
    fused_fusion_w1<<<NAVT, 256, 0, stream>>>(audio, vision, text, W1, y1p,
                                              ones, zeros);
    norm_kernel    <<<B_,   256, 0, stream>>>(audio, vision, text, normbuf);
    mlp_tail       <<<B_,   128, 0, stream>>>(y1p, b1, W2, b2, W3, b3, out);
    reg_reduce     <<<1,    32,  0, stream>>>(normbuf, out);
}